// MGDCFEncoder_56169582297147
// MI455X (gfx1250) — compile-verified
//
#include <hip/hip_runtime.h>
#include <hip/hip_bf16.h>
#include <stdint.h>

#define EMB_D     64
#define TILE_E    1024
#define SPMM_TPB  256
#define SPMM_WAVES (SPMM_TPB / 32)
#define CPK       (TILE_E / SPMM_TPB)   // elements per thread per array per tile (=4)
#define ALPHA_F   0.1f
#define BETA_F    0.9f

// ---- CDNA5 async global->LDS path (guarded; fallback = synchronous copy) ----
#if defined(__HIP_DEVICE_COMPILE__) && \
    __has_builtin(__builtin_amdgcn_global_load_async_to_lds_b32) && \
    __has_builtin(__builtin_amdgcn_s_wait_asynccnt)
#define HAVE_ASYNC_LDS 1
#else
#define HAVE_ASYNC_LDS 0
#endif

#define GLOBAL_AS __attribute__((address_space(1)))
#define LDS_AS    __attribute__((address_space(3)))

__device__ __forceinline__ void async_cp_b32(const void* gsrc, void* ldst) {
#if HAVE_ASYNC_LDS
  __builtin_amdgcn_global_load_async_to_lds_b32(
      (GLOBAL_AS int*)gsrc, (LDS_AS int*)ldst, 0, 0);
#else
  *(uint32_t*)ldst = *(const uint32_t*)gsrc;
#endif
}

#if HAVE_ASYNC_LDS
#define ASYNC_WAIT(n) __builtin_amdgcn_s_wait_asynccnt(n)
#else
#define ASYNC_WAIT(n) ((void)0)
#endif

// ---- SpMM: y[r*64+d] += v * x[c*64+d] over COO edges, double-buffered LDS tiles ----
__global__ __launch_bounds__(SPMM_TPB)
void spmm_kernel(const int* __restrict__ rows, const int* __restrict__ cols,
                 const float* __restrict__ vals, const float* __restrict__ x,
                 float* __restrict__ y, int E, int ntiles) {
  __shared__ int   s_rows[2][TILE_E];
  __shared__ int   s_cols[2][TILE_E];
  __shared__ float s_vals[2][TILE_E];

  const int tid  = threadIdx.x;
  const int wave = tid >> 5;
  const int lane = tid & 31;

  int t = (int)blockIdx.x;
  if (t >= ntiles) return;

  auto load_tile = [&](int tt, int b) {
    const int base = tt * TILE_E;
#pragma unroll
    for (int k = 0; k < CPK; ++k) {
      const int j = k * SPMM_TPB + tid;
      int g = base + j;
      if (g > E - 1) g = E - 1;  // clamp: always issue -> deterministic ASYNCcnt
      async_cp_b32(rows + g, &s_rows[b][j]);
      async_cp_b32(cols + g, &s_cols[b][j]);
      async_cp_b32(vals + g, &s_vals[b][j]);
    }
  };

  int buf = 0;
  load_tile(t, buf);

  for (; t < ntiles; t += (int)gridDim.x) {
    const int tn = t + (int)gridDim.x;
    if (tn < ntiles) {
      load_tile(tn, buf ^ 1);      // prefetch next tile into other half
      ASYNC_WAIT(3 * CPK);         // current tile's 12 copies retired (in-order)
    } else {
      ASYNC_WAIT(0);
    }
    __syncthreads();

    const int base = t * TILE_E;
    int count = E - base;
    if (count > TILE_E) count = TILE_E;

    const int*   sr = s_rows[buf];
    const int*   sc = s_cols[buf];
    const float* sv = s_vals[buf];

    // one edge per wave; lane handles a float2 of the 64-wide embedding row
    for (int e = wave; e < count; e += SPMM_WAVES) {
      const int   r = sr[e];           // same-address LDS broadcast
      const int   c = sc[e];
      const float v = sv[e];
      const float2 xv = *(const float2*)(x + (size_t)c * EMB_D + 2 * lane);
      float* yp = y + (size_t)r * EMB_D + 2 * lane;
      atomicAdd(yp,     v * xv.x);     // global_atomic_add_f32, no return
      atomicAdd(yp + 1, v * xv.y);
    }
    __syncthreads();   // all waves done reading before this half is reloaded
    buf ^= 1;
  }
}

// ---- final fused 3-term combine + 1/gamma scale + batched row gather ----
// out = c3*y3[row] + c2*y2[row] + c1*x0[row]
__global__ void gather_kernel(const float* __restrict__ y3, const float* __restrict__ y2,
                              const float* __restrict__ x0,
                              const int* __restrict__ users, const int* __restrict__ items,
                              float* __restrict__ out, int B, int U,
                              float c3, float c2, float c1) {
  const int t = (int)(blockIdx.x * blockDim.x + threadIdx.x);
  const int total = 2 * B * EMB_D;
  if (t >= total) return;
  const int part = t / (B * EMB_D);        // 0 = users, 1 = items
  const int rem  = t - part * (B * EMB_D);
  const int b    = rem >> 6;
  const int d    = rem & (EMB_D - 1);
  const int row  = part ? (U + items[b]) : users[b];
  const size_t off = (size_t)row * EMB_D + d;
  out[t] = c3 * y3[off] + c2 * y2[off] + c1 * x0[off];
}

extern "C" void kernel_launch(void* const* d_in, const int* in_sizes, int n_in,
                              void* d_out, int out_size, void* d_ws, size_t ws_size,
                              hipStream_t stream) {
  const int*   users    = (const int*)d_in[0];
  const int*   items    = (const int*)d_in[1];
  const float* user_emb = (const float*)d_in[2];
  const float* item_emb = (const float*)d_in[3];
  const int*   adj_rows = (const int*)d_in[4];
  const int*   adj_cols = (const int*)d_in[5];
  const float* adj_vals = (const float*)d_in[6];

  const int B = in_sizes[0];
  const int U = in_sizes[2] / EMB_D;
  const int I = in_sizes[3] / EMB_D;
  const int E = in_sizes[4];
  const int N = U + I;
  const size_t ND = (size_t)N * EMB_D;

  float* buf0 = (float*)d_ws;       // ego0, then y3 = A@y2
  float* buf1 = buf0 + ND;          // x0 = ego1 = A@ego0
  float* buf2 = buf1 + ND;          // y2 = A@x0 (uncombined)

  // ego0 = concat(user_emb, item_emb)
  (void)hipMemcpyAsync(buf0, user_emb, (size_t)U * EMB_D * sizeof(float),
                       hipMemcpyDeviceToDevice, stream);
  (void)hipMemcpyAsync(buf0 + (size_t)U * EMB_D, item_emb,
                       (size_t)I * EMB_D * sizeof(float),
                       hipMemcpyDeviceToDevice, stream);

  const int ntiles = (E + TILE_E - 1) / TILE_E;
  const int blocks = ntiles < 512 ? ntiles : 512;

  // x0 = A @ ego0
  (void)hipMemsetAsync(buf1, 0, ND * sizeof(float), stream);
  spmm_kernel<<<blocks, SPMM_TPB, 0, stream>>>(adj_rows, adj_cols, adj_vals,
                                               buf0, buf1, E, ntiles);

  // y2 = A @ x0  (combine deferred via linearity: A@(b*y2+a*x0) = b*A@y2 + a*y2)
  (void)hipMemsetAsync(buf2, 0, ND * sizeof(float), stream);
  spmm_kernel<<<blocks, SPMM_TPB, 0, stream>>>(adj_rows, adj_cols, adj_vals,
                                               buf1, buf2, E, ntiles);

  // y3 = A @ y2
  (void)hipMemsetAsync(buf0, 0, ND * sizeof(float), stream);
  spmm_kernel<<<blocks, SPMM_TPB, 0, stream>>>(adj_rows, adj_cols, adj_vals,
                                               buf2, buf0, E, ntiles);

  // ego3/gamma = [b^2*y3 + a*b*y2 + a*x0] / gamma,  gamma = b^3 + a*(1+b+b^2)
  float bp = 1.0f, gamma = 0.0f;
  for (int i = 0; i < 3; ++i) { gamma += ALPHA_F * bp; bp *= BETA_F; }
  gamma += bp;
  const float inv_g = 1.0f / gamma;
  const float c3 = BETA_F * BETA_F * inv_g;
  const float c2 = ALPHA_F * BETA_F * inv_g;
  const float c1 = ALPHA_F * inv_g;

  const int total = 2 * B * EMB_D;
  gather_kernel<<<(total + 255) / 256, 256, 0, stream>>>(
      buf0, buf2, buf1, users, items, (float*)d_out, B, U, c3, c2, c1);
}